// CategoricalCrossentropy_32908039422195
// MI455X (gfx1250) — compile-verified
//
#include <hip/hip_runtime.h>
#include <stdint.h>

// Problem geometry (fixed by the reference)
#define N_ROWS     65536
#define C_COLS     1024
#define HALF_COLS  512           // process the matrix in two 128MB column halves (L2 reuse)
#define CHUNK_ROWS 128
#define NCHUNK     (N_ROWS / CHUNK_ROWS)      // 512 row-chunks
#define TILE_ROWS  16
#define TILE_COLS  256
#define NSUB       (CHUNK_ROWS / TILE_ROWS)   // 8 subtiles per chunk
#define BT         256                         // threads per block (8 waves on wave32)
#define CHUNKS_PER_TILE ((TILE_ROWS * TILE_COLS * 4) / 16 / BT)  // 4 x 16B per thread
#define LOG2E      1.44269504088896340736f
#define EPSF       1e-9f

#if defined(__has_builtin)
#  if __has_builtin(__builtin_amdgcn_global_load_async_to_lds_b128)
#    define HAS_ASYNC_LDS 1
#  endif
#endif
#ifndef HAS_ASYNC_LDS
#  define HAS_ASYNC_LDS 0
#endif

// Pointer types the async builtin wants: int4 pointee (16B), LDS side in AS(3).
typedef int v4i __attribute__((vector_size(16)));
typedef __attribute__((address_space(3))) v4i as3_v4i;

// Copy a TILE_ROWS x TILE_COLS f32 tile (row stride C_COLS in global) into LDS.
// Async path: GLOBAL_LOAD_ASYNC_TO_LDS_B128, 4 in-flight per thread (ASYNCcnt).
__device__ __forceinline__ void tile_copy(const float* __restrict__ g, float* l, int tid)
{
#if HAS_ASYNC_LDS
#pragma unroll
    for (int i = 0; i < CHUNKS_PER_TILE; ++i) {
        const int k   = tid + i * BT;
        const int row = k >> 6;      // 64 x 16B chunks per 256-float row
        const int cc  = k & 63;
        __builtin_amdgcn_global_load_async_to_lds_b128(
            (v4i*)(g + (size_t)row * C_COLS + (size_t)cc * 4),
            (as3_v4i*)(l + row * TILE_COLS + cc * 4),
            0, 0);
    }
#else
#pragma unroll
    for (int i = 0; i < CHUNKS_PER_TILE; ++i) {
        const int k   = tid + i * BT;
        const int row = k >> 6;
        const int cc  = k & 63;
        const float4 v = *(const float4*)(g + (size_t)row * C_COLS + (size_t)cc * 4);
        *(float4*)(l + row * TILE_COLS + cc * 4) = v;
    }
#endif
}

template <int PENDING>
__device__ __forceinline__ void tile_wait()
{
#if HAS_ASYNC_LDS
#  if __has_builtin(__builtin_amdgcn_s_wait_asynccnt)
    __builtin_amdgcn_s_wait_asynccnt(PENDING);
#  else
    if constexpr (PENDING == 0)
        asm volatile("s_wait_asynccnt 0x0" ::: "memory");
    else
        asm volatile("s_wait_asynccnt 0x4" ::: "memory");   // == CHUNKS_PER_TILE
#  endif
#endif
}

// ---------------------------------------------------------------------------
// Pass 1: per-(chunk, column) online softmax stats in base-2.
// Thread t owns column col0+t; y = x*log2(e); keeps running max m and s=Σ2^(y-m).
// ---------------------------------------------------------------------------
__global__ __launch_bounds__(BT) void stats_kernel(
    const float* __restrict__ pred,
    float* __restrict__ pmax, float* __restrict__ psum, int colBase)
{
    __shared__ float tile[2][TILE_ROWS * TILE_COLS];   // 32 KB double buffer
    const int tid     = threadIdx.x;
    const int col0    = colBase + blockIdx.x * TILE_COLS;
    const int rowBase = blockIdx.y * CHUNK_ROWS;

    float m = -3.0e38f, s = 0.0f;

    tile_copy(pred + (size_t)rowBase * C_COLS + col0, tile[0], tid);
#pragma unroll 1
    for (int sub = 0; sub < NSUB; ++sub) {
        const int cur = sub & 1;
        if (sub + 1 < NSUB) {
            tile_copy(pred + (size_t)(rowBase + (sub + 1) * TILE_ROWS) * C_COLS + col0,
                      tile[1 - cur], tid);
            tile_wait<CHUNKS_PER_TILE>();   // current tile landed, next still in flight
        } else {
            tile_wait<0>();
        }
        __syncthreads();

        const float* t = tile[cur];
        float msub = -3.0e38f;
#pragma unroll
        for (int r = 0; r < TILE_ROWS; ++r)
            msub = fmaxf(msub, t[r * TILE_COLS + tid]);
        const float mNew = fmaxf(m, msub * LOG2E);
        s *= __builtin_amdgcn_exp2f(m - mNew);          // rescale once per subtile
        float acc = 0.0f;
#pragma unroll
        for (int r = 0; r < TILE_ROWS; ++r)
            acc += __builtin_amdgcn_exp2f(
                       __builtin_fmaf(t[r * TILE_COLS + tid], LOG2E, -mNew));
        s += acc;
        m  = mNew;
        __syncthreads();                                // buffer reuse fence
    }
    const size_t o = (size_t)blockIdx.y * HALF_COLS + (blockIdx.x * TILE_COLS + tid);
    pmax[o] = m;
    psum[o] = s;
}

// ---------------------------------------------------------------------------
// Combine chunk partials -> per-column M (base-2 max) and 1/S.
// ---------------------------------------------------------------------------
__global__ __launch_bounds__(BT) void combine_kernel(
    const float* __restrict__ pmax, const float* __restrict__ psum,
    float* __restrict__ colM, float* __restrict__ colRS, int colBase)
{
    const int cl = blockIdx.x * BT + threadIdx.x;       // 0..HALF_COLS-1
    float M = -3.0e38f;
    for (int k = 0; k < NCHUNK; ++k)
        M = fmaxf(M, pmax[(size_t)k * HALF_COLS + cl]);
    float S = 0.0f;
    for (int k = 0; k < NCHUNK; ++k)
        S += psum[(size_t)k * HALF_COLS + cl] *
             __builtin_amdgcn_exp2f(pmax[(size_t)k * HALF_COLS + cl] - M);
    colM[colBase + cl]  = M;
    colRS[colBase + cl] = 1.0f / S;
}

// ---------------------------------------------------------------------------
// Pass 2: Σ log2-BCE terms. p = 2^(y - M)·(1/S) + eps; target element uses
// log2(p), all others log2(1-p). Targets staged once per chunk into LDS.
// ---------------------------------------------------------------------------
__global__ __launch_bounds__(BT) void loss_kernel(
    const float* __restrict__ pred,
    const long long* __restrict__ target,
    const float* __restrict__ colM, const float* __restrict__ colRS,
    float* __restrict__ partial, int colBase)
{
    __shared__ float tile[2][TILE_ROWS * TILE_COLS];
    __shared__ int   tgt[CHUNK_ROWS];
    __shared__ float red[BT];
    const int tid     = threadIdx.x;
    const int col0    = colBase + blockIdx.x * TILE_COLS;
    const int c       = col0 + tid;
    const int rowBase = blockIdx.y * CHUNK_ROWS;

    for (int i = tid; i < CHUNK_ROWS; i += BT)          // 512B of targets
        tgt[i] = (int)target[rowBase + i];

    const float M  = colM[c];
    const float RS = colRS[c];
    float acc = 0.0f;

    tile_copy(pred + (size_t)rowBase * C_COLS + col0, tile[0], tid);
#pragma unroll 1
    for (int sub = 0; sub < NSUB; ++sub) {
        const int cur = sub & 1;
        if (sub + 1 < NSUB) {
            tile_copy(pred + (size_t)(rowBase + (sub + 1) * TILE_ROWS) * C_COLS + col0,
                      tile[1 - cur], tid);
            tile_wait<CHUNKS_PER_TILE>();
        } else {
            tile_wait<0>();
        }
        __syncthreads();                                // also fences tgt staging (sub==0)

        const float* t = tile[cur];
#pragma unroll
        for (int r = 0; r < TILE_ROWS; ++r) {
            const float x = t[r * TILE_COLS + tid];
            const float e = __builtin_amdgcn_exp2f(__builtin_fmaf(x, LOG2E, -M));
            const float p = __builtin_fmaf(e, RS, EPSF);
            const float q = (tgt[sub * TILE_ROWS + r] == c) ? p : (1.0f - p);
            acc += __builtin_amdgcn_logf(q);            // v_log_f32 is log2
        }
        __syncthreads();
    }

    red[tid] = acc;
    __syncthreads();
#pragma unroll
    for (int off = BT / 2; off > 0; off >>= 1) {
        if (tid < off) red[tid] += red[tid + off];
        __syncthreads();
    }
    if (tid == 0)
        partial[blockIdx.y * gridDim.x + blockIdx.x] = red[0];
}

__global__ __launch_bounds__(BT) void finalize_kernel(
    const float* __restrict__ partial, int n, float* __restrict__ out)
{
    __shared__ float red[BT];
    float a = 0.0f;
    for (int i = threadIdx.x; i < n; i += BT) a += partial[i];
    red[threadIdx.x] = a;
    __syncthreads();
#pragma unroll
    for (int off = BT / 2; off > 0; off >>= 1) {
        if (threadIdx.x < off) red[threadIdx.x] += red[threadIdx.x + off];
        __syncthreads();
    }
    if (threadIdx.x == 0)
        out[0] = -red[0] * (1.0f / ((float)N_ROWS * (float)C_COLS)); // exact 2^-26
}

extern "C" void kernel_launch(void* const* d_in, const int* in_sizes, int n_in,
                              void* d_out, int out_size, void* d_ws, size_t ws_size,
                              hipStream_t stream)
{
    (void)in_sizes; (void)n_in; (void)out_size; (void)ws_size;
    const float*     pred   = (const float*)d_in[0];
    const long long* target = (const long long*)d_in[1];
    float*           out    = (float*)d_out;

    // Workspace carve-up (~2.02 MB total), all regions fully overwritten each call.
    float* pmaxW   = (float*)d_ws;                          // NCHUNK*HALF_COLS
    float* psumW   = pmaxW + (size_t)NCHUNK * HALF_COLS;    // NCHUNK*HALF_COLS
    float* colM    = psumW + (size_t)NCHUNK * HALF_COLS;    // C_COLS
    float* colRS   = colM + C_COLS;                         // C_COLS
    float* partial = colRS + C_COLS;                        // 2 * (2*NCHUNK)

    const dim3 grid(HALF_COLS / TILE_COLS, NCHUNK);         // (2, 512) blocks per half

    // Two column halves: each half's pass-2 re-reads 128 MB that pass-1 just
    // pulled through the 192 MB L2.
    for (int h = 0; h < 2; ++h) {
        stats_kernel  <<<grid, BT, 0, stream>>>(pred, pmaxW, psumW, h * HALF_COLS);
        combine_kernel<<<dim3(HALF_COLS / BT), BT, 0, stream>>>(pmaxW, psumW, colM, colRS,
                                                                h * HALF_COLS);
        loss_kernel   <<<grid, BT, 0, stream>>>(pred, target, colM, colRS,
                                                partial + (size_t)h * (2 * NCHUNK),
                                                h * HALF_COLS);
    }
    finalize_kernel<<<1, BT, 0, stream>>>(partial, 2 * (2 * NCHUNK), out);
}